// StackedDynamicGraphNeuralNetwork_33741263077528
// MI455X (gfx1250) — compile-verified
//
#include <hip/hip_runtime.h>
#include <hip/hip_bf16.h>

typedef float v2f __attribute__((ext_vector_type(2)));
typedef float v8f __attribute__((ext_vector_type(8)));

#define BATCH 16
#define NN    2048
#define DF    64

// -------------------------------------------------------------------------
// t[b][d] = sum_n x[b][n][d]
__global__ __launch_bounds__(256) void colsum_k(const float* __restrict__ x,
                                                float* __restrict__ tsum) {
    __shared__ float sd[256];
    const int b = blockIdx.x;
    const int tid = threadIdx.x;
    const int d = tid & 63;
    const int part = tid >> 6;            // 0..3
    const float* xb = x + (size_t)b * NN * DF;
    float s = 0.0f;
    for (int n = part; n < NN; n += 4)
        s += xb[(size_t)n * DF + d];
    sd[tid] = s;
    __syncthreads();
    if (tid < DF)
        tsum[b * DF + tid] = sd[tid] + sd[64 + tid] + sd[128 + tid] + sd[192 + tid];
}

// -------------------------------------------------------------------------
// dinv[row] = rsqrt(x[row]·t[b] + 1)  (guard rowsum==0 -> 1), one wave/row
__global__ __launch_bounds__(256) void dinv_k(const float* __restrict__ x,
                                              const float* __restrict__ tsum,
                                              float* __restrict__ dinv) {
    const int wave = threadIdx.x >> 5;
    const int lane = threadIdx.x & 31;
    const int grow = blockIdx.x * 8 + wave;      // global row, 0..B*NN-1
    const int b = grow >> 11;                    // NN == 2048
    const float* xr = x + (size_t)grow * DF;
    const float* tb = tsum + b * DF;
    float v = xr[lane] * tb[lane] + xr[lane + 32] * tb[lane + 32];
#pragma unroll
    for (int off = 16; off > 0; off >>= 1)
        v += __shfl_xor(v, off, 32);
    const float rs = v + 1.0f;
    const float di = (rs == 0.0f) ? 1.0f : rsqrtf(rs);
    if (lane == 0) dinv[grow] = di;
}

// -------------------------------------------------------------------------
// Hs[b][n][f] = dinv[n] * ( (X W)[n][f] + bias[f] ), WMMA f32 16x16x4.
// Block: 256 thr = 8 waves, each wave one 16-row strip. Grid (NN/128, B).
__global__ __launch_bounds__(256) void hs_k(const float* __restrict__ x,
                                            const float* __restrict__ W,
                                            const float* __restrict__ bias,
                                            const float* __restrict__ dinv,
                                            float* __restrict__ Hs) {
    __shared__ float Ws[DF * DF];
    __shared__ float bs[DF];
    const int b = blockIdx.y;
    const int tid = threadIdx.x;
    for (int i = tid; i < DF * DF; i += 256) Ws[i] = W[i];
    if (tid < DF) bs[tid] = bias[tid];
    __syncthreads();

    const int wave = tid >> 5, lane = tid & 31;
    const int r = lane & 15, h = lane >> 4;
    const int n0 = blockIdx.x * 128 + wave * 16;
    const float* xrow = x + ((size_t)b * NN + n0 + r) * DF;

    v8f c0 = {}, c1 = {}, c2 = {}, c3 = {};
#pragma unroll
    for (int k0 = 0; k0 < 16; ++k0) {
        const int kb = k0 * 4 + 2 * h;
        v2f a;  a.x = xrow[kb];  a.y = xrow[kb + 1];
        v2f bv;
        bv.x = Ws[kb * DF +  0 + r]; bv.y = Ws[(kb + 1) * DF +  0 + r];
        c0 = __builtin_amdgcn_wmma_f32_16x16x4_f32(false, a, false, bv, (short)0, c0, false, false);
        bv.x = Ws[kb * DF + 16 + r]; bv.y = Ws[(kb + 1) * DF + 16 + r];
        c1 = __builtin_amdgcn_wmma_f32_16x16x4_f32(false, a, false, bv, (short)0, c1, false, false);
        bv.x = Ws[kb * DF + 32 + r]; bv.y = Ws[(kb + 1) * DF + 32 + r];
        c2 = __builtin_amdgcn_wmma_f32_16x16x4_f32(false, a, false, bv, (short)0, c2, false, false);
        bv.x = Ws[kb * DF + 48 + r]; bv.y = Ws[(kb + 1) * DF + 48 + r];
        c3 = __builtin_amdgcn_wmma_f32_16x16x4_f32(false, a, false, bv, (short)0, c3, false, false);
    }

    const float* dv = dinv + (size_t)b * NN + n0;
    float* hrow = Hs + ((size_t)b * NN + n0) * DF;
#pragma unroll
    for (int v = 0; v < 8; ++v) {
        const int m = v + 8 * h;
        const float dm = dv[m];
        hrow[m * DF +  0 + r] = dm * (c0[v] + bs[ 0 + r]);
        hrow[m * DF + 16 + r] = dm * (c1[v] + bs[16 + r]);
        hrow[m * DF + 32 + r] = dm * (c2[v] + bs[32 + r]);
        hrow[m * DF + 48 + r] = dm * (c3[v] + bs[48 + r]);
    }
}

// -------------------------------------------------------------------------
// G[b] = X[b]^T @ Hs[b]   (64x64, reduce over n=2048). One block per batch;
// 8 waves x 2 tiles cover the 4x4 grid of 16x16 output tiles.
__global__ __launch_bounds__(256) void g_k(const float* __restrict__ x,
                                           const float* __restrict__ Hs,
                                           float* __restrict__ G) {
    const int b = blockIdx.x;
    const int tid = threadIdx.x;
    const int wave = tid >> 5, lane = tid & 31;
    const int r = lane & 15, h = lane >> 4;
    const int mt  = wave >> 1;           // tile row 0..3  (d dimension)
    const int ntb = (wave & 1) * 2;      // tile cols {0,1} or {2,3}

    const float* xb = x  + (size_t)b * NN * DF;
    const float* hb = Hs + (size_t)b * NN * DF;

    v8f c0 = {}, c1 = {};
    for (int k = 0; k < NN / 4; ++k) {
        const int kb = k * 4 + 2 * h;
        v2f a;
        a.x = xb[(size_t)kb * DF + mt * 16 + r];           // A[m=d][k=n] = x[n][d]
        a.y = xb[(size_t)(kb + 1) * DF + mt * 16 + r];
        v2f b0, b1;
        b0.x = hb[(size_t)kb * DF + ntb * 16 + r];
        b0.y = hb[(size_t)(kb + 1) * DF + ntb * 16 + r];
        b1.x = hb[(size_t)kb * DF + (ntb + 1) * 16 + r];
        b1.y = hb[(size_t)(kb + 1) * DF + (ntb + 1) * 16 + r];
        c0 = __builtin_amdgcn_wmma_f32_16x16x4_f32(false, a, false, b0, (short)0, c0, false, false);
        c1 = __builtin_amdgcn_wmma_f32_16x16x4_f32(false, a, false, b1, (short)0, c1, false, false);
    }

    float* Gb = G + (size_t)b * DF * DF;
#pragma unroll
    for (int v = 0; v < 8; ++v) {
        const int m = mt * 16 + v + 8 * h;
        Gb[m * DF + ntb * 16 + r]       = c0[v];
        Gb[m * DF + (ntb + 1) * 16 + r] = c1[v];
    }
}

// -------------------------------------------------------------------------
// out[b][n][f] = relu( dinv[n] * ( (X G)[n][f] + Hs[n][f] ) )
__global__ __launch_bounds__(256) void out_k(const float* __restrict__ x,
                                             const float* __restrict__ G,
                                             const float* __restrict__ Hs,
                                             const float* __restrict__ dinv,
                                             float* __restrict__ out) {
    __shared__ float Gs[DF * DF];
    const int b = blockIdx.y;
    const int tid = threadIdx.x;
    for (int i = tid; i < DF * DF; i += 256) Gs[i] = G[(size_t)b * DF * DF + i];
    __syncthreads();

    const int wave = tid >> 5, lane = tid & 31;
    const int r = lane & 15, h = lane >> 4;
    const int n0 = blockIdx.x * 128 + wave * 16;
    const float* xrow = x + ((size_t)b * NN + n0 + r) * DF;

    v8f c0 = {}, c1 = {}, c2 = {}, c3 = {};
#pragma unroll
    for (int k0 = 0; k0 < 16; ++k0) {
        const int kb = k0 * 4 + 2 * h;
        v2f a;  a.x = xrow[kb];  a.y = xrow[kb + 1];
        v2f bv;
        bv.x = Gs[kb * DF +  0 + r]; bv.y = Gs[(kb + 1) * DF +  0 + r];
        c0 = __builtin_amdgcn_wmma_f32_16x16x4_f32(false, a, false, bv, (short)0, c0, false, false);
        bv.x = Gs[kb * DF + 16 + r]; bv.y = Gs[(kb + 1) * DF + 16 + r];
        c1 = __builtin_amdgcn_wmma_f32_16x16x4_f32(false, a, false, bv, (short)0, c1, false, false);
        bv.x = Gs[kb * DF + 32 + r]; bv.y = Gs[(kb + 1) * DF + 32 + r];
        c2 = __builtin_amdgcn_wmma_f32_16x16x4_f32(false, a, false, bv, (short)0, c2, false, false);
        bv.x = Gs[kb * DF + 48 + r]; bv.y = Gs[(kb + 1) * DF + 48 + r];
        c3 = __builtin_amdgcn_wmma_f32_16x16x4_f32(false, a, false, bv, (short)0, c3, false, false);
    }

    const float* dv = dinv + (size_t)b * NN + n0;
    const float* hrow = Hs + ((size_t)b * NN + n0) * DF;
    float* orow = out + ((size_t)b * NN + n0) * DF;
#pragma unroll
    for (int v = 0; v < 8; ++v) {
        const int m = v + 8 * h;
        const float dm = dv[m];
        orow[m * DF +  0 + r] = fmaxf(dm * (c0[v] + hrow[m * DF +  0 + r]), 0.0f);
        orow[m * DF + 16 + r] = fmaxf(dm * (c1[v] + hrow[m * DF + 16 + r]), 0.0f);
        orow[m * DF + 32 + r] = fmaxf(dm * (c2[v] + hrow[m * DF + 32 + r]), 0.0f);
        orow[m * DF + 48 + r] = fmaxf(dm * (c3[v] + hrow[m * DF + 48 + r]), 0.0f);
    }
}

// -------------------------------------------------------------------------
extern "C" void kernel_launch(void* const* d_in, const int* in_sizes, int n_in,
                              void* d_out, int out_size, void* d_ws, size_t ws_size,
                              hipStream_t stream) {
    (void)in_sizes; (void)n_in; (void)out_size; (void)ws_size;

    const float* x = (const float*)d_in[0];
    const float* W[3]    = {(const float*)d_in[1], (const float*)d_in[3], (const float*)d_in[5]};
    const float* bias[3] = {(const float*)d_in[2], (const float*)d_in[4], (const float*)d_in[6]};
    float* out = (float*)d_out;

    const size_t BN64 = (size_t)BATCH * NN * DF;   // 2,097,152 floats
    float* ws   = (float*)d_ws;
    float* bufA = ws;                  // layer-0 output
    float* bufB = bufA + BN64;         // layer-1 output
    float* Hs   = bufB + BN64;         // scaled hidden
    float* dinv = Hs + BN64;           // B*NN
    float* tsum = dinv + (size_t)BATCH * NN;   // B*64
    float* G    = tsum + (size_t)BATCH * DF;   // B*64*64

    const float* lin[3]  = {x, bufA, bufB};
    float*       lout[3] = {bufA, bufB, out};

    for (int l = 0; l < 3; ++l) {
        colsum_k<<<BATCH, 256, 0, stream>>>(lin[l], tsum);
        dinv_k<<<(BATCH * NN) / 8, 256, 0, stream>>>(lin[l], tsum, dinv);
        hs_k<<<dim3(NN / 128, BATCH), 256, 0, stream>>>(lin[l], W[l], bias[l], dinv, Hs);
        g_k<<<BATCH, 256, 0, stream>>>(lin[l], Hs, G);
        out_k<<<dim3(NN / 128, BATCH), 256, 0, stream>>>(lin[l], G, Hs, dinv, lout[l]);
    }
}